// SpecAugment_46136538694073
// MI455X (gfx1250) — compile-verified
//
#include <hip/hip_runtime.h>
#include <stdint.h>

// ---------------------------------------------------------------------------
// SpecAugment for MI455X (gfx1250, wave32).
// Memory-bound: ~328 MB traffic -> ~14 us at 23.3 TB/s. Strategy: stream rows
// with B128 accesses; stage input through LDS with the CDNA5 async-DMA path
// (global_load_async_to_lds_b128 + s_wait_asynccnt double buffering); skip
// reads entirely for fully-masked frequency rows.
// ---------------------------------------------------------------------------

#define B_  128
#define F_  80
#define T_  4000
#define FREQ_MASKS 2
#define TIME_MASKS 10
#define FREQ_SPAN  28   // (MAX_FREQ+1) - MIN_FREQ = 27+1
#define TIME_SPAN  101  // (MAX_TIME+1) - MIN_TIME = 100+1

#define NV  1000        // float4 vectors per row (4000 / 4)
#define CH  256         // float4 vectors per staged chunk (= blockDim)
#define NCHUNK 4        // ceil(1000/256)

// ------------------------- JAX threefry-2x32 PRNG --------------------------

__device__ __forceinline__ uint32_t rotl32(uint32_t x, int r) {
  return (x << r) | (x >> (32 - r));
}

__device__ __forceinline__ void threefry2x32(uint32_t k0, uint32_t k1,
                                             uint32_t c0, uint32_t c1,
                                             uint32_t& o0, uint32_t& o1) {
  uint32_t ks0 = k0, ks1 = k1, ks2 = 0x1BD11BDAu ^ k0 ^ k1;
  uint32_t x0 = c0 + ks0, x1 = c1 + ks1;
#define QR(r) x0 += x1; x1 = rotl32(x1, r); x1 ^= x0;
  QR(13) QR(15) QR(26) QR(6)   x0 += ks1; x1 += ks2 + 1u;
  QR(17) QR(29) QR(16) QR(24)  x0 += ks2; x1 += ks0 + 2u;
  QR(13) QR(15) QR(26) QR(6)   x0 += ks0; x1 += ks1 + 3u;
  QR(17) QR(29) QR(16) QR(24)  x0 += ks1; x1 += ks2 + 4u;
  QR(13) QR(15) QR(26) QR(6)   x0 += ks2; x1 += ks0 + 5u;
#undef QR
  o0 = x0; o1 = x1;
}

// jax.random.split(key) -> 2 keys (counts [0,1,2,3], paired as (0,2),(1,3))
__device__ __forceinline__ void jsplit(uint32_t k0, uint32_t k1,
                                       uint32_t& a0, uint32_t& a1,
                                       uint32_t& b0, uint32_t& b1) {
  uint32_t y00, y01, y10, y11;
  threefry2x32(k0, k1, 0u, 2u, y00, y01);
  threefry2x32(k0, k1, 1u, 3u, y10, y11);
  a0 = y00; a1 = y10;   // key0 = (out[0], out[1])
  b0 = y01; b1 = y11;   // key1 = (out[2], out[3])
}

// Element j of an N-element (N even) 32-bit random_bits draw.
__device__ __forceinline__ uint32_t rbits(uint32_t k0, uint32_t k1,
                                          uint32_t N, uint32_t j) {
  uint32_t half = N >> 1;
  uint32_t p = (j < half) ? j : j - half;
  uint32_t o0, o1;
  threefry2x32(k0, k1, p, p + half, o0, o1);
  return (j < half) ? o0 : o1;
}

// jax.random.randint with minval=0 (squared-multiplier mod-span formula).
__device__ __forceinline__ int jrandint(uint32_t bits, uint32_t span) {
  uint32_t hi = bits >> 16, lo = bits & 0xFFFFu;
  uint32_t mult = 65536u % span;
  mult = (mult * mult) % span;
  uint32_t off = ((hi % span) * mult + (lo % span)) % span;
  return (int)off;
}

// jax.random.uniform in [0,1): bitcast((bits>>9)|0x3f800000) - 1.0f
__device__ __forceinline__ float juniform(uint32_t bits) {
  uint32_t u = (bits >> 9) | 0x3F800000u;
  return __uint_as_float(u) - 1.0f;
}

// ------------------------- mask-parameter kernel ---------------------------
// One thread per batch sample. Writes 24 ints per sample into d_ws:
//   [0..3]   : freq start0, end0, start1, end1
//   [4..13]  : time starts (10)
//   [14..23] : time ends   (10)
__global__ void sa_params_kernel(int* __restrict__ P) {
  int b = threadIdx.x;
  if (b >= B_) return;

  uint32_t kf0, kf1, kt0, kt1;
  jsplit(0u, 42u, kf0, kf1, kt0, kt1);   // root key(42) = (0, 42)

  uint32_t kw0, kw1, ks0, ks1;

  // frequency bands
  jsplit(kf0, kf1, kw0, kw1, ks0, ks1);
  for (int m = 0; m < FREQ_MASKS; ++m) {
    uint32_t j = (uint32_t)(b * FREQ_MASKS + m);
    int   w  = jrandint(rbits(kw0, kw1, B_ * FREQ_MASKS, j), FREQ_SPAN);
    float u  = juniform(rbits(ks0, ks1, B_ * FREQ_MASKS, j));
    int   hi = F_ - w; if (hi < 1) hi = 1;
    int   s  = (int)(u * (float)hi);
    P[b * 24 + 2 * m]     = s;
    P[b * 24 + 2 * m + 1] = s + w;
  }

  // time bands
  jsplit(kt0, kt1, kw0, kw1, ks0, ks1);
  for (int m = 0; m < TIME_MASKS; ++m) {
    uint32_t j = (uint32_t)(b * TIME_MASKS + m);
    int   w  = jrandint(rbits(kw0, kw1, B_ * TIME_MASKS, j), TIME_SPAN);
    float u  = juniform(rbits(ks0, ks1, B_ * TIME_MASKS, j));
    int   hi = T_ - w; if (hi < 1) hi = 1;
    int   s  = (int)(u * (float)hi);
    P[b * 24 + 4 + m]  = s;
    P[b * 24 + 14 + m] = s + w;
  }
}

// ------------------------------ apply kernel -------------------------------
// One block per (b, f) row of 4000 floats. 256 threads = 8 wave32s.
__global__ __launch_bounds__(256) void sa_apply_kernel(
    const float* __restrict__ x, float* __restrict__ y,
    const int* __restrict__ P) {
  const int f   = blockIdx.x;
  const int b   = blockIdx.y;
  const int tid = threadIdx.x;

  __shared__ int      sp[24];
  __shared__ uint32_t tmask[125];        // 4000-bit time mask
  __shared__ float4   buf[2][CH];        // async-DMA double buffer (8 KB)

  if (tid < 24) sp[tid] = P[b * 24 + tid];
  __syncthreads();

  // Build the time-mask bitset with interval arithmetic (no per-bit loop).
  if (tid < 125) {
    const int base = tid * 32;
    uint32_t m = 0;
    #pragma unroll
    for (int i = 0; i < TIME_MASKS; ++i) {
      int lo = sp[4 + i]  - base;  if (lo < 0)  lo = 0;
      int hi = sp[14 + i] - base;  if (hi > 32) hi = 32;
      if (hi > lo) {
        uint32_t himask = (hi >= 32) ? 0xFFFFFFFFu : ((1u << hi) - 1u);
        uint32_t lomask = (lo <= 0)  ? 0u          : ((1u << lo) - 1u);
        m |= himask & ~lomask;
      }
    }
    tmask[tid] = m;
  }
  const bool freq_masked =
      (f >= sp[0] && f < sp[1]) || (f >= sp[2] && f < sp[3]);
  __syncthreads();

  const size_t row = ((size_t)b * F_ + f) * (size_t)T_;
  float4* __restrict__ out4 = (float4*)(y + row);

  if (freq_masked) {
    // Whole row zero: write-only path (saves ~1/3 of total read bandwidth).
    const float4 z = make_float4(0.f, 0.f, 0.f, 0.f);
    for (int i = tid; i < NV; i += 256) out4[i] = z;
    return;
  }

  const float4* __restrict__ in4 = (const float4*)(x + row);

  // LDS byte offsets of this lane's staging slots (generic addr low 32 bits).
  const uint32_t lds0 = (uint32_t)(size_t)&buf[0][tid];
  const uint32_t lds1 = (uint32_t)(size_t)&buf[1][tid];

  // Prologue: async-DMA chunk 0 into LDS (per-lane B128, tracked by ASYNCcnt).
  {
    int i = tid; if (i > NV - 1) i = NV - 1;  // clamp tail lanes in-bounds
    unsigned long long ga = (unsigned long long)(const void*)(in4 + i);
    asm volatile("global_load_async_to_lds_b128 %0, %1, off"
                 :: "v"(lds0), "v"(ga) : "memory");
  }

  #pragma unroll
  for (int c = 0; c < NCHUNK; ++c) {
    if (c + 1 < NCHUNK) {
      // Prefetch chunk c+1, then wait until chunk c has landed (loads are
      // in-order, so ASYNCcnt <= 1 means the older transfer completed).
      int i = (c + 1) * CH + tid; if (i > NV - 1) i = NV - 1;
      unsigned long long ga = (unsigned long long)(const void*)(in4 + i);
      const uint32_t ldsn = ((c + 1) & 1) ? lds1 : lds0;
      asm volatile("global_load_async_to_lds_b128 %0, %1, off"
                   :: "v"(ldsn), "v"(ga) : "memory");
      asm volatile("s_wait_asynccnt 1" ::: "memory");
    } else {
      asm volatile("s_wait_asynccnt 0" ::: "memory");
    }

    // Each lane consumes only the LDS slot its own async op wrote -> no
    // block barrier needed; per-wave ASYNCcnt wait is sufficient.
    float4 v = buf[c & 1][tid];

    const int i  = c * CH + tid;
    int im = i; if (im > NV - 1) im = NV - 1;
    const int t0 = im * 4;                        // multiple of 4: nibble
    const uint32_t nib = (tmask[t0 >> 5] >> (t0 & 31)) & 0xFu;
    v.x = (nib & 1u) ? 0.f : v.x;
    v.y = (nib & 2u) ? 0.f : v.y;
    v.z = (nib & 4u) ? 0.f : v.z;
    v.w = (nib & 8u) ? 0.f : v.w;

    if (i < NV) out4[i] = v;
  }
}

// ------------------------------- launcher ----------------------------------
extern "C" void kernel_launch(void* const* d_in, const int* in_sizes, int n_in,
                              void* d_out, int out_size, void* d_ws,
                              size_t ws_size, hipStream_t stream) {
  (void)in_sizes; (void)n_in; (void)out_size; (void)ws_size;
  const float* x = (const float*)d_in[0];
  float*       y = (float*)d_out;
  int*         P = (int*)d_ws;   // 128 * 24 ints = 12 KB of scratch

  sa_params_kernel<<<dim3(1), dim3(128), 0, stream>>>(P);
  sa_apply_kernel<<<dim3(F_, B_), dim3(256), 0, stream>>>(x, y, P);
}